// FLAME_5669356832384
// MI455X (gfx1250) — compile-verified
//
#include <hip/hip_runtime.h>
#include <math.h>

typedef __attribute__((ext_vector_type(2))) float v2f;
typedef __attribute__((ext_vector_type(8))) float v8f;

#define B_    512
#define V_    5023
#define NJ    5
#define NBETA 400
#define M3    15069           // V_*3
#define P_    36              // (NJ-1)*9
#define NTILE 942             // ceil(M3/16)
#define LDT   108             // padded LDS row stride: 16B-aligned rows (108%4==0),
                              // 108*4 mod 64 banks = 44, gcd(44,64)=4 -> 16 lanes hit
                              // 16 distinct bank-pairs for b64 reads (conflict-free)

// workspace layout (floats)
#define WS_VS 0                       // v_shaped / v_posed : B_*M3
#define WS_PF (WS_VS + B_*M3)         // pose_feature       : B_*P_
#define WS_A  (WS_PF + B_*P_)         // A transforms       : B_*NJ*16
#define WS_J  (WS_A  + B_*NJ*16)     // joints J           : B_*NJ*3

#if defined(__AMDGCN__) && __has_builtin(__builtin_amdgcn_global_load_async_to_lds_b128)
#define HAVE_ASYNC_LDS 1
#define AS1 __attribute__((address_space(1)))
#define AS3 __attribute__((address_space(3)))
typedef int v4i_ __attribute__((vector_size(16)));   // builtin's pointee type
__device__ __forceinline__ void async_copy16(const float* gsrc, float* ldst) {
    __builtin_amdgcn_global_load_async_to_lds_b128(
        (AS1 v4i_*)const_cast<float*>(gsrc),
        (AS3 v4i_*)ldst, 0, 0);
}
__device__ __forceinline__ void wait_async_zero() {
#if __has_builtin(__builtin_amdgcn_s_wait_asynccnt)
    __builtin_amdgcn_s_wait_asynccnt(0);
#else
    asm volatile("s_wait_asynccnt 0x0" ::: "memory");
#endif
}
#endif

// ---------------------------------------------------------------------------
// K1: v_shaped[b][n] = v_template[n] + sum_k betas[b][k] * shapedirs[n][k]
//     C(512 x 15069) = A(512 x 400) * B^T(400 x 15069), f32 WMMA 16x16x4.
//     8 waves/block: one 16-row M tile shared, 8 adjacent 16-col N tiles.
//     Staging: GLOBAL_LOAD_ASYNC_TO_LDS_B128 (ASYNCcnt) when available.
// ---------------------------------------------------------------------------
__global__ __launch_bounds__(256)
void k1_shape_gemm(const float* __restrict__ betas,
                   const float* __restrict__ shapedirs,
                   const float* __restrict__ v_template,
                   float* __restrict__ vs) {
    __shared__ float As[16 * LDT];        // betas tile chunk (16 x 100)
    __shared__ float Bs[8 * 16 * LDT];    // 8 shapedirs tiles (16 x 100 each)

    const int tid   = threadIdx.x;
    const int lane  = tid & 31;
    const int w     = tid >> 5;
    const int lm    = lane & 15;
    const int half  = lane >> 4;          // 0: K pair {0,1}; 1: K pair {2,3}
    const int mtile = blockIdx.y;
    const int ntile = blockIdx.x * 8 + w;

    v8f acc = {};
    for (int kc = 0; kc < 4; ++kc) {
        const int k0 = kc * 100;
        __syncthreads();  // previous chunk's compute done before overwriting LDS
#if defined(HAVE_ASYNC_LDS)
        // A tile: 16 rows x 25 float4 chunks, direct global->LDS async copies
        for (int e = tid; e < 16 * 25; e += 256) {
            int r = e / 25, c4 = e - r * 25;
            async_copy16(&betas[(mtile * 16 + r) * NBETA + k0 + c4 * 4],
                         &As[r * LDT + c4 * 4]);
        }
        // B tiles: 8 x 16 rows x 25 float4 chunks; clamp edge rows
        for (int e = tid; e < 8 * 16 * 25; e += 256) {
            int t = e / 400;
            int rem = e - t * 400;
            int r = rem / 25, c4 = rem - r * 25;
            int g = (blockIdx.x * 8 + t) * 16 + r;
            g = g < M3 ? g : (M3 - 1);
            async_copy16(&shapedirs[g * NBETA + k0 + c4 * 4],
                         &Bs[t * (16 * LDT) + r * LDT + c4 * 4]);
            // prefetch next K-chunk, ~one hint per 128B line
            if (kc < 3 && (c4 & 7) == 0)
                __builtin_prefetch(&shapedirs[g * NBETA + k0 + 100 + c4 * 4], 0, 1);
        }
        wait_async_zero();   // our LDS writes landed; barrier makes them visible
#else
        // fallback: synchronous staging through VGPRs
        for (int e = tid; e < 16 * 100; e += 256) {
            int r = e / 100, c = e - r * 100;
            As[r * LDT + c] = betas[(mtile * 16 + r) * NBETA + k0 + c];
        }
        for (int e = tid; e < 8 * 16 * 100; e += 256) {
            int t = e / 1600;
            int rem = e - t * 1600;
            int r = rem / 100, c = rem - r * 100;
            int g = (blockIdx.x * 8 + t) * 16 + r;
            g = g < M3 ? g : (M3 - 1);
            Bs[t * (16 * LDT) + r * LDT + c] = shapedirs[g * NBETA + k0 + c];
            if (kc < 3 && (c & 31) == 0)
                __builtin_prefetch(&shapedirs[g * NBETA + k0 + 100 + c], 0, 1);
        }
#endif
        __syncthreads();

        const float* Bw = &Bs[w * (16 * LDT)];
        #pragma unroll
        for (int kk = 0; kk < 25; ++kk) {
            const int kb = kk * 4 + 2 * half;
            v2f a, b;
            // A frag: lanes 0-15 (M=lm) hold K={kb,kb+1}; lanes 16-31 K={kb,kb+1} with kb=kk*4+2
            a.x = As[lm * LDT + kb];
            a.y = As[lm * LDT + kb + 1];
            // B frag (4x16, K x N): same pairing, N=lm within this wave's tile
            b.x = Bw[lm * LDT + kb];
            b.y = Bw[lm * LDT + kb + 1];
            acc = __builtin_amdgcn_wmma_f32_16x16x4_f32(
                false, a, false, b, (short)0, acc, false, false);
        }
    }
    // C layout: VGPR r -> (M = r + 8*half, N = lm)
    const int n = ntile * 16 + lm;
    if (ntile < NTILE && n < M3) {
        const float vt = v_template[n];
        const int mb = mtile * 16 + half * 8;
        #pragma unroll
        for (int r = 0; r < 8; ++r)
            vs[(mb + r) * M3 + n] = acc[r] + vt;
    }
}

// ---------------------------------------------------------------------------
// K2: J[b][j][k] = sum_v J_regressor[j][v] * v_shaped[b][v*3+k]
//     one wave32 per output, butterfly reduce
// ---------------------------------------------------------------------------
__global__ __launch_bounds__(256)
void k2_jreg(const float* __restrict__ JR,
             const float* __restrict__ vs,
             float* __restrict__ Jout) {
    const int lane = threadIdx.x & 31;
    const int gid  = blockIdx.x * 8 + (threadIdx.x >> 5);
    if (gid >= B_ * NJ * 3) return;
    const int b   = gid / (NJ * 3);
    const int rem = gid - b * (NJ * 3);
    const int j   = rem / 3;
    const int k   = rem - j * 3;
    float s = 0.0f;
    for (int v = lane; v < V_; v += 32)
        s += JR[j * V_ + v] * vs[b * M3 + v * 3 + k];
    #pragma unroll
    for (int off = 16; off > 0; off >>= 1)
        s += __shfl_xor(s, off, 32);
    if (lane == 0) Jout[gid] = s;
}

// ---------------------------------------------------------------------------
// K3: Rodrigues + pose_feature + kinematic chain. Thread per batch.
// ---------------------------------------------------------------------------
__global__ __launch_bounds__(256)
void k3_chain(const float* __restrict__ pose,
              const int* __restrict__ parents,
              const float* __restrict__ Jws,
              float* __restrict__ pf,
              float* __restrict__ Aws,
              float* __restrict__ Jt_out) {
    const int b = blockIdx.x * blockDim.x + threadIdx.x;
    if (b >= B_) return;

    float R[NJ][9], Jl[NJ][3];
    #pragma unroll
    for (int j = 0; j < NJ; ++j) {
        float x = pose[b * 15 + j * 3 + 0];
        float y = pose[b * 15 + j * 3 + 1];
        float z = pose[b * 15 + j * 3 + 2];
        float ax = x + 1e-8f, ay = y + 1e-8f, az = z + 1e-8f;
        float ang = sqrtf(ax * ax + ay * ay + az * az);
        float inv = 1.0f / ang;
        float rx = x * inv, ry = y * inv, rz = z * inv;
        float c = cosf(ang), s = sinf(ang), o = 1.0f - c;
        R[j][0] = 1.0f - o * (ry * ry + rz * rz);
        R[j][1] = -s * rz + o * rx * ry;
        R[j][2] =  s * ry + o * rx * rz;
        R[j][3] =  s * rz + o * rx * ry;
        R[j][4] = 1.0f - o * (rx * rx + rz * rz);
        R[j][5] = -s * rx + o * ry * rz;
        R[j][6] = -s * ry + o * rx * rz;
        R[j][7] =  s * rx + o * ry * rz;
        R[j][8] = 1.0f - o * (rx * rx + ry * ry);
        Jl[j][0] = Jws[b * 15 + j * 3 + 0];
        Jl[j][1] = Jws[b * 15 + j * 3 + 1];
        Jl[j][2] = Jws[b * 15 + j * 3 + 2];
    }
    // pose_feature = (R[1:] - I), row-major
    for (int j = 1; j < NJ; ++j)
        for (int i = 0; i < 9; ++i)
            pf[b * P_ + (j - 1) * 9 + i] = R[j][i] - ((i % 4 == 0) ? 1.0f : 0.0f);

    // chain composition (3x3 rot + translation; bottom row implicit)
    float cR[NJ][9], ct[NJ][3];
    for (int i = 0; i < 9; ++i) cR[0][i] = R[0][i];
    for (int k = 0; k < 3; ++k) ct[0][k] = Jl[0][k];
    for (int j = 1; j < NJ; ++j) {
        int p = parents[j];
        float rel[3] = { Jl[j][0] - Jl[p][0], Jl[j][1] - Jl[p][1], Jl[j][2] - Jl[p][2] };
        for (int r = 0; r < 3; ++r) {
            for (int cc = 0; cc < 3; ++cc)
                cR[j][r * 3 + cc] = cR[p][r * 3 + 0] * R[j][0 * 3 + cc]
                                  + cR[p][r * 3 + 1] * R[j][1 * 3 + cc]
                                  + cR[p][r * 3 + 2] * R[j][2 * 3 + cc];
            ct[j][r] = cR[p][r * 3 + 0] * rel[0] + cR[p][r * 3 + 1] * rel[1]
                     + cR[p][r * 3 + 2] * rel[2] + ct[p][r];
        }
    }
    // posed joints + relative transforms A = [cR | ct - cR*J ; 0 0 0 1]
    for (int j = 0; j < NJ; ++j) {
        for (int k = 0; k < 3; ++k)
            Jt_out[b * 15 + j * 3 + k] = ct[j][k];
        for (int r = 0; r < 3; ++r) {
            float tj = cR[j][r * 3 + 0] * Jl[j][0] + cR[j][r * 3 + 1] * Jl[j][1]
                     + cR[j][r * 3 + 2] * Jl[j][2];
            Aws[b * (NJ * 16) + j * 16 + r * 4 + 0] = cR[j][r * 3 + 0];
            Aws[b * (NJ * 16) + j * 16 + r * 4 + 1] = cR[j][r * 3 + 1];
            Aws[b * (NJ * 16) + j * 16 + r * 4 + 2] = cR[j][r * 3 + 2];
            Aws[b * (NJ * 16) + j * 16 + r * 4 + 3] = ct[j][r] - tj;
        }
        Aws[b * (NJ * 16) + j * 16 + 12] = 0.0f;
        Aws[b * (NJ * 16) + j * 16 + 13] = 0.0f;
        Aws[b * (NJ * 16) + j * 16 + 14] = 0.0f;
        Aws[b * (NJ * 16) + j * 16 + 15] = 1.0f;
    }
}

// ---------------------------------------------------------------------------
// K4a: vs += pose_feature(512x36) @ posedirs(36x15069). WMMA, K=36 -> 9 steps.
//      posedirs is K-major, so B-fragment loads coalesce across lanes.
// ---------------------------------------------------------------------------
__global__ __launch_bounds__(256)
void k4a_pose_gemm(const float* __restrict__ pf,
                   const float* __restrict__ posedirs,
                   float* __restrict__ vs) {
    const int lane  = threadIdx.x & 31;
    const int w     = threadIdx.x >> 5;
    const int lm    = lane & 15;
    const int half  = lane >> 4;
    const int mtile = blockIdx.y;
    const int ntile = blockIdx.x * 8 + w;
    const int n  = ntile * 16 + lm;
    const int nc = n < M3 ? n : (M3 - 1);
    const int m  = mtile * 16 + lm;

    v8f acc = {};
    #pragma unroll
    for (int kk = 0; kk < 9; ++kk) {
        const int kb = kk * 4 + 2 * half;
        v2f a, b;
        a.x = pf[m * P_ + kb];
        a.y = pf[m * P_ + kb + 1];
        b.x = posedirs[kb * M3 + nc];
        b.y = posedirs[(kb + 1) * M3 + nc];
        acc = __builtin_amdgcn_wmma_f32_16x16x4_f32(
            false, a, false, b, (short)0, acc, false, false);
    }
    if (ntile < NTILE && n < M3) {
        const int mb = mtile * 16 + half * 8;
        #pragma unroll
        for (int r = 0; r < 8; ++r)
            vs[(mb + r) * M3 + n] += acc[r];   // v_posed in place
    }
}

// ---------------------------------------------------------------------------
// K4b: verts = (sum_j w[v][j] * A[b][j]) * [v_posed, 1]
// ---------------------------------------------------------------------------
__global__ __launch_bounds__(256)
void k4b_lbs(const float* __restrict__ Aws,
             const float* __restrict__ Wl,
             const float* __restrict__ vs,
             float* __restrict__ verts) {
    __shared__ float sA[NJ * 16];
    const int b = blockIdx.y;
    if (threadIdx.x < NJ * 16)
        sA[threadIdx.x] = Aws[b * (NJ * 16) + threadIdx.x];
    __syncthreads();
    const int v = blockIdx.x * 256 + threadIdx.x;
    if (v >= V_) return;
    float wj[NJ];
    #pragma unroll
    for (int j = 0; j < NJ; ++j) wj[j] = Wl[v * NJ + j];
    float T[12];
    #pragma unroll
    for (int i = 0; i < 12; ++i) {
        float t = 0.0f;
        #pragma unroll
        for (int j = 0; j < NJ; ++j) t += wj[j] * sA[j * 16 + i];
        T[i] = t;
    }
    const float px = vs[b * M3 + v * 3 + 0];
    const float py = vs[b * M3 + v * 3 + 1];
    const float pz = vs[b * M3 + v * 3 + 2];
    verts[(b * V_ + v) * 3 + 0] = T[0] * px + T[1] * py + T[2]  * pz + T[3];
    verts[(b * V_ + v) * 3 + 1] = T[4] * px + T[5] * py + T[6]  * pz + T[7];
    verts[(b * V_ + v) * 3 + 2] = T[8] * px + T[9] * py + T[10] * pz + T[11];
}

extern "C" void kernel_launch(void* const* d_in, const int* in_sizes, int n_in,
                              void* d_out, int out_size, void* d_ws, size_t ws_size,
                              hipStream_t stream) {
    const float* betas      = (const float*)d_in[0];
    const float* pose       = (const float*)d_in[1];
    const float* v_template = (const float*)d_in[2];
    const float* shapedirs  = (const float*)d_in[3];
    const float* posedirs   = (const float*)d_in[4];
    const float* JR         = (const float*)d_in[5];
    const float* Wl         = (const float*)d_in[6];
    const int*   parents    = (const int*)d_in[7];

    float* out   = (float*)d_out;
    float* ws    = (float*)d_ws;
    float* vs    = ws + WS_VS;
    float* pf    = ws + WS_PF;
    float* Aws   = ws + WS_A;
    float* Jws   = ws + WS_J;
    float* verts = out;
    float* Jt    = out + (size_t)B_ * V_ * 3;

    const dim3 blk(256);
    const dim3 gGemm(118, 32);   // 118*8 >= 942 N tiles, 32 M tiles

    k1_shape_gemm<<<gGemm, blk, 0, stream>>>(betas, shapedirs, v_template, vs);
    k2_jreg<<<dim3((B_ * NJ * 3 + 7) / 8), blk, 0, stream>>>(JR, vs, Jws);
    k3_chain<<<dim3((B_ + 255) / 256), blk, 0, stream>>>(pose, parents, Jws, pf, Aws, Jt);
    k4a_pose_gemm<<<gGemm, blk, 0, stream>>>(pf, posedirs, vs);
    k4b_lbs<<<dim3((V_ + 255) / 256, B_), blk, 0, stream>>>(Aws, Wl, vs, verts);
}